// SigmoidRT_45406394253470
// MI455X (gfx1250) — compile-verified
//
#include <hip/hip_runtime.h>
#include <stdint.h>

// ---------------------------------------------------------------------------
// out[b,n,m,f] = e0 + e1 * sigmoid((z[b,n,m,f] - e2) * e3),
//   (e0,e1,e2,e3) = eta_fault[Mask[b,f]]
//
// z [8,64,512,256] fp32 (256 MiB) -> out (256 MiB). Pure stream:
// roofline floor = 512 MiB / 23.3 TB/s ~= 23 us. No contraction -> WMMA
// inapplicable. CDNA5 path used instead: GLOBAL_LOAD_ASYNC_TO_LDS_B128
// (ASYNCcnt-tracked DMA into per-wave double buffers) so in-flight bytes are
// not limited by VGPRs, + non-temporal b128 stores so the 512 MiB stream
// doesn't thrash the 192 MB L2.
// ---------------------------------------------------------------------------

typedef float f32x4 __attribute__((ext_vector_type(4)));

#define B_ 8
#define N_ 64
#define M_ 512
#define F_ 256
#define THREADS 256
#define WAVES 8                 // wave32
#define LANES 32
#define VEC_PER_LANE 4          // async b128 loads per lane per chunk
#define CHUNK_VEC (LANES * VEC_PER_LANE)            // 128 vec4 = 2 KB / chunk
#define CHUNKS 8                                    // chunks per wave
#define VEC_PER_BLOCK (WAVES * CHUNKS * CHUNK_VEC)  // 8192 vec4 per block
#define VEC_PER_BATCH ((N_ * M_ * F_) / 4)          // 2,097,152
#define BLOCKS_PER_BATCH (VEC_PER_BATCH / VEC_PER_BLOCK) // 256

struct SMem {
  f32x4 eta[F_];                    // 4 KB  : eta row per f (this batch)
  f32x4 zbuf[WAVES][2][CHUNK_VEC];  // 32 KB : per-wave double buffers
};

// Raw LDS byte offset of a __shared__ object (addrspace(3) ptrtoint).
__device__ __forceinline__ unsigned lds_off(const void* p) {
  return (unsigned)(uint64_t)(__attribute__((address_space(3))) const char*)p;
}

// One chunk = 4 async b128 per lane. INST_OFFSET advances BOTH the global
// and the LDS address (CDNA5 ISA), so slot j lives at
// lds_base + j*512 and global_base + j*512 simultaneously.
__device__ __forceinline__ void issue_chunk(unsigned ldsa, unsigned goff,
                                            uint64_t base) {
  asm volatile(
      "global_load_async_to_lds_b128 %0, %1, %2 th:TH_LOAD_NT\n\t"
      "global_load_async_to_lds_b128 %0, %1, %2 offset:512 th:TH_LOAD_NT\n\t"
      "global_load_async_to_lds_b128 %0, %1, %2 offset:1024 th:TH_LOAD_NT\n\t"
      "global_load_async_to_lds_b128 %0, %1, %2 offset:1536 th:TH_LOAD_NT"
      :: "v"(ldsa), "v"(goff), "s"(base) : "memory");
}

__device__ __forceinline__ float sig_eval(float z, float a, float c,
                                          float e0, float e1) {
  // sigmoid((z-e2)*e3) = 1/(1+exp2(z*a + c)), a=-log2e*e3, c=log2e*e2*e3
  float ex = __builtin_amdgcn_exp2f(__builtin_fmaf(z, a, c)); // v_exp_f32
  float s  = __builtin_amdgcn_rcpf(1.0f + ex);                // v_rcp_f32
  return __builtin_fmaf(e1, s, e0);                           // v_fma_f32
}

__global__ void __launch_bounds__(THREADS)
sigmoid_rt_kernel(const float* __restrict__ z,
                  const long long* __restrict__ mask,
                  const float* __restrict__ eta_fault,
                  float* __restrict__ out) {
  __shared__ SMem smem;

  const int tid  = threadIdx.x;
  const int lane = tid & (LANES - 1);
  const int w    = tid >> 5;
  const int b    = blockIdx.y;

  const long long bbase = (long long)b * (long long)(N_ * M_ * F_);
  const uint64_t  zb    = (uint64_t)(const void*)(z + bbase);

  // This wave's base vec4 index within the batch (mod 64 == 0 by design).
  const unsigned wavevec =
      (unsigned)blockIdx.x * VEC_PER_BLOCK + (unsigned)w * (CHUNKS * CHUNK_VEC);

  // LDS byte addresses of this wave's two buffer phases (+ this lane's slot).
  const unsigned lb0 = lds_off(&smem.zbuf[w][0][0]) + lane * 16;
  const unsigned lb1 = lds_off(&smem.zbuf[w][1][0]) + lane * 16;
  // Global byte offset of this lane in chunk 0.
  const unsigned gb = (wavevec + (unsigned)lane) * 16u;

  // Prime the pipeline: chunks 0 and 1 in flight while we gather eta.
  issue_chunk(lb0, gb + 0u * (CHUNK_VEC * 16), zb);
  issue_chunk(lb1, gb + 1u * (CHUNK_VEC * 16), zb);

  // Gather eta_fault[Mask[b,f]] into LDS (one 16B row per thread).
  {
    const long long idx = mask[(long long)b * F_ + tid]; // int64, 0..14
    smem.eta[tid] = ((const f32x4*)eta_fault)[idx];
  }
  __syncthreads();

  // Slot j of this lane always maps to v mod 64 = lane + 32*(j&1), i.e.
  // f-rows 4*(lane+32h)+k. Hoist those 8 eta rows into folded constants.
  float e0[2][4], e1[2][4], aa[2][4], cc[2][4];
  const float L2E = 1.4426950408889634f;
#pragma unroll
  for (int h = 0; h < 2; ++h) {
#pragma unroll
    for (int k = 0; k < 4; ++k) {
      f32x4 e = smem.eta[(lane + 32 * h) * 4 + k];
      e0[h][k] = e.x;
      e1[h][k] = e.y;
      aa[h][k] = -L2E * e.w;
      cc[h][k] = L2E * e.z * e.w;
    }
  }

  f32x4* op = (f32x4*)(out + bbase);

#pragma unroll
  for (int c = 0; c < CHUNKS; ++c) {
    const int ph = c & 1;
    // Async loads retire in order: <=4 outstanding => chunk c has landed
    // (only chunk c+1's 4 ops may remain). Last chunk: drain fully.
    if (c < CHUNKS - 1) asm volatile("s_wait_asynccnt 4" ::: "memory");
    else                asm volatile("s_wait_asynccnt 0" ::: "memory");

    const unsigned cvec = wavevec + (unsigned)c * CHUNK_VEC;
#pragma unroll
    for (int j = 0; j < 4; ++j) {
      f32x4 zz = smem.zbuf[w][ph][j * 32 + lane]; // ds_load_b128
      const int h = j & 1;
      f32x4 r;
      r.x = sig_eval(zz.x, aa[h][0], cc[h][0], e0[h][0], e1[h][0]);
      r.y = sig_eval(zz.y, aa[h][1], cc[h][1], e0[h][1], e1[h][1]);
      r.z = sig_eval(zz.z, aa[h][2], cc[h][2], e0[h][2], e1[h][2]);
      r.w = sig_eval(zz.w, aa[h][3], cc[h][3], e0[h][3], e1[h][3]);
      // Streamed once: non-temporal b128 store.
      __builtin_nontemporal_store(r, &op[cvec + (unsigned)(j * 32 + lane)]);
    }

    // Refill the phase we just consumed with chunk c+2.
    const int nc = c + 2;
    if (nc < CHUNKS)
      issue_chunk(ph == 0 ? lb0 : lb1, gb + (unsigned)nc * (CHUNK_VEC * 16), zb);
  }
}

extern "C" void kernel_launch(void* const* d_in, const int* in_sizes, int n_in,
                              void* d_out, int out_size, void* d_ws,
                              size_t ws_size, hipStream_t stream) {
  (void)in_sizes; (void)n_in; (void)out_size; (void)d_ws; (void)ws_size;
  const float*     z    = (const float*)d_in[0];
  const long long* mask = (const long long*)d_in[1];  // int64 per reference
  const float*     eta  = (const float*)d_in[2];
  float*           out  = (float*)d_out;

  dim3 grid(BLOCKS_PER_BATCH, B_, 1); // 256 x 8 blocks, 256 thr (8 wave32)
  sigmoid_rt_kernel<<<grid, THREADS, 0, stream>>>(z, mask, eta, out);
}